// ERCGNN_19662360281516
// MI455X (gfx1250) — compile-verified
//
#include <hip/hip_runtime.h>

// ---------------- problem constants (match reference) ----------------
#define NNODES 50000
#define NEDGES 500000
#define DD     128
#define NH     8
#define DHH    16
#define NLAY   2
#define NCLS   7
#define BN_EPS 1e-9f

#define LDSP   132   // padded LDS row stride (132 % 64 == 4 -> conflict-free b64 reads)

typedef __attribute__((ext_vector_type(2))) float v2f;
typedef __attribute__((ext_vector_type(8))) float v8f;

// ---------------------------------------------------------------------------
// Tiled GEMM: C[M x ncols] (+)= A[M x 128] @ W (+ bias), optional ReLU.
// One block per 16-row M tile; WAVES waves, wave w owns N-tile n0 = 16*w.
// A tile staged in LDS (coalesced float4), fragments via ds_load_2addr_b64.
// fp32 WMMA 16x16x4 fragment layouts (ISA 7.12.2):
//   A 16x4 : lane l: M=l&15, VGPR0=K(2*(l>>4)), VGPR1=K(2*(l>>4)+1)
//   B 4x16 : lane l: N=l&15, VGPR g holds K = g + 2*(l>>4)
//   C 16x16: lane l: N=l&15, VGPR g holds M = g + 8*(l>>4)
// All strides are template constants so B/C addresses fold into the 24-bit
// instruction offsets (no per-iteration 64-bit address adds).
// ---------------------------------------------------------------------------
template<int WAVES, int LDC, bool WHEAD, bool NMASK, bool ACCUM, bool BIAS, bool RELU>
__global__ __launch_bounds__(WAVES * 32)
void gemm16(const float* __restrict__ A,
            const float* __restrict__ W,
            const float* __restrict__ bias,
            float* __restrict__ C,
            int ncols)
{
    __shared__ float As[16 * LDSP];

    const int tid = threadIdx.x;
    const int m0  = blockIdx.x * 16;

    // ---- stage A tile: 16 rows x 128 cols, coalesced float4 loads ----
    for (int idx = tid; idx < 16 * 32; idx += WAVES * 32) {
        const int r = idx >> 5, c4 = idx & 31;
        *(float4*)&As[r * LDSP + c4 * 4] =
            *(const float4*)(A + (long)(m0 + r) * DD + c4 * 4);
    }
    __syncthreads();

    const int wv   = tid >> 5;
    const int lane = tid & 31;
    const int half = lane >> 4;      // 0/1
    const int li   = lane & 15;
    const int n0   = wv * 16;
    const int n    = n0 + li;
    const bool nok = NMASK ? (n < ncols) : true;
    const int  nc  = NMASK ? (nok ? n : 0) : n;   // clamped (keeps loads in-bounds)

    v8f c;
#pragma unroll
    for (int g = 0; g < 8; ++g) c[g] = 0.0f;
    if (ACCUM) {
#pragma unroll
        for (int g = 0; g < 8; ++g) {
            const int m = m0 + g + 8 * half;
            float v = C[(long)m * LDC + nc];
            if (NMASK && !nok) v = 0.0f;
            c[g] = v;
        }
    }

    // B fragment base: lane holds K = k0 + 2*half (VGPR0) and +1 (VGPR1).
    constexpr int BS = WHEAD ? DHH : LDC;   // weight stride along K
    const float* bp = WHEAD
        ? (W + ((long)(nc >> 4) * DD + 2 * half) * DHH + (nc & 15))
        : (W + (long)(2 * half) * BS + nc);

    const float* ap = &As[li * LDSP + 2 * half];

#pragma unroll 16
    for (int k0 = 0; k0 < DD; k0 += 4) {
        v2f a = *(const v2f*)(ap + k0);             // ds_load_b64, conflict-free
        float bx = bp[(long)k0 * BS];               // immediate-offset loads
        float by = bp[(long)k0 * BS + BS];
        if (NMASK && !nok) { bx = 0.0f; by = 0.0f; }
        v2f b; b.x = bx; b.y = by;
        c = __builtin_amdgcn_wmma_f32_16x16x4_f32(
                false, a, false, b, (short)0, c, false, false);
    }

    float bv = 0.0f;
    if (BIAS) bv = (NMASK && !nok) ? 0.0f : bias[nc];

    float* cp = C + (long)m0 * LDC + n;
#pragma unroll
    for (int g = 0; g < 8; ++g) {
        const int m = g + 8 * half;
        float v = c[g] + bv;
        if (RELU) v = fmaxf(v, 0.0f);
        if (!NMASK || nok) cp[(long)m * LDC] = v;   // immediate-offset stores
    }
}

// ---------------------------------------------------------------------------
// Per-edge scatter-add. One thread per (edge, float4 chunk); 32 chunks = 128.
//   GAT: w = (ats[row][h] + atn[col][h]) * vals[e];  GCN: w = vals[e]
// agg[row] += w * hsrc[col]
// ---------------------------------------------------------------------------
template<bool GAT>
__global__ void scatter_edges(const float* __restrict__ hsrc,
                              const float* __restrict__ ats,
                              const float* __restrict__ atn,
                              const int* __restrict__ row,
                              const int* __restrict__ col,
                              const float* __restrict__ vals,
                              float* __restrict__ agg, int E)
{
    const int t = blockIdx.x * blockDim.x + threadIdx.x;
    const int e = t >> 5;
    if (e >= E) return;
    const int q  = t & 31;
    const int r  = row[e];
    const int cc = col[e];
    float w = vals[e];
    if (GAT) {
        const int h = q >> 2;
        w *= (ats[r * NH + h] + atn[cc * NH + h]);
    }
    const float4 hv = *(const float4*)(hsrc + (long)cc * DD + q * 4);
    float* dst = agg + (long)r * DD + q * 4;
    atomicAdd(dst + 0, w * hv.x);
    atomicAdd(dst + 1, w * hv.y);
    atomicAdd(dst + 2, w * hv.z);
    atomicAdd(dst + 3, w * hv.w);
}

// Attention logits: ats/atn[n][h] = leaky_relu(dot(hs[n,h,:], a[h,:]), 0.2)
// (faithful to reference: BOTH use hs)
__global__ void gat_att(const float* __restrict__ hs,
                        const float* __restrict__ a_s,
                        const float* __restrict__ a_n,
                        float* __restrict__ ats, float* __restrict__ atn, int N)
{
    const int t = blockIdx.x * blockDim.x + threadIdx.x;
    if (t >= N * NH) return;
    const int n = t / NH, h = t % NH;
    const float* hp = hs + (long)n * DD + h * DHH;
    float s0 = 0.0f, s1 = 0.0f;
#pragma unroll
    for (int k = 0; k < DHH; ++k) {
        const float v = hp[k];
        s0 += v * a_s[h * DHH + k];
        s1 += v * a_n[h * DHH + k];
    }
    ats[t] = (s0 > 0.0f) ? s0 : 0.2f * s0;
    atn[t] = (s1 > 0.0f) ? s1 : 0.2f * s1;
}

__global__ void fillf(float* __restrict__ p, long n, float v)
{
    const long i = (long)blockIdx.x * blockDim.x + threadIdx.x;
    if (i < n) p[i] = v;
}

// BN pass 1: per-column sum / sumsq (thread j owns column j; coalesced rows)
__global__ void bn_stats(const float* __restrict__ x, float* __restrict__ sums,
                         int N, int prerelu)
{
    const int j = threadIdx.x;             // 0..127
    float s = 0.0f, s2 = 0.0f;
    for (int r = blockIdx.x; r < N; r += gridDim.x) {
        float v = x[(long)r * DD + j];
        if (prerelu) v = fmaxf(v, 0.0f);
        s  += v;
        s2 += v * v;
    }
    atomicAdd(&sums[j], s);
    atomicAdd(&sums[DD + j], s2);
}

// BN pass 2: sums -> (mean, rstd)
__global__ void bn_finalize(float* __restrict__ stats, int N)
{
    const int j = threadIdx.x;
    if (j >= DD) return;
    const float inv = 1.0f / (float)N;
    const float m   = stats[j] * inv;
    const float var = stats[DD + j] * inv - m * m;   // population var (ddof=0)
    stats[j]      = m;
    stats[DD + j] = rsqrtf(var + BN_EPS);
}

// BN pass 3: y = (relu?(x) - mean) * rstd
__global__ void bn_apply(const float* __restrict__ x, float* __restrict__ y,
                         const float* __restrict__ stats, long n, int prerelu)
{
    const long i = (long)blockIdx.x * blockDim.x + threadIdx.x;
    if (i >= n) return;
    const int j = (int)(i & (DD - 1));
    float v = x[i];
    if (prerelu) v = fmaxf(v, 0.0f);
    y[i] = (v - stats[j]) * stats[DD + j];
}

// ---------------------------------------------------------------------------
// host-side launch helpers
// ---------------------------------------------------------------------------
static void gemm_full(const float* A, const float* W, const float* bias, float* C,
                      bool relu, hipStream_t s)
{
    // standard weights W[k*DD + n], 128 output cols, bias, optional relu
    if (relu)
        gemm16<8, DD, false, false, false, true, true ><<<NNODES / 16, 256, 0, s>>>(A, W, bias, C, DD);
    else
        gemm16<8, DD, false, false, false, true, false><<<NNODES / 16, 256, 0, s>>>(A, W, bias, C, DD);
}

static void gemm_head(const float* A, const float* W, const float* bias, float* C,
                      hipStream_t s)
{
    // GAT per-head weights W[(h*128+k)*16+j], bias+relu
    gemm16<8, DD, true, false, false, true, true><<<NNODES / 16, 256, 0, s>>>(A, W, bias, C, DD);
}

static void gemm_cls(const float* A, const float* W, const float* bias, float* C,
                     bool first, hipStream_t s)
{
    if (first)  // C = A@W + bias
        gemm16<1, NCLS, false, true, false, true,  false><<<NNODES / 16, 32, 0, s>>>(A, W, bias, C, NCLS);
    else        // C += A@W
        gemm16<1, NCLS, false, true, true,  false, false><<<NNODES / 16, 32, 0, s>>>(A, W, nullptr, C, NCLS);
}

static void run_bn(const float* x, float* y, float* stats, int prerelu, hipStream_t s)
{
    fillf<<<1, 256, 0, s>>>(stats, 2 * DD, 0.0f);
    bn_stats<<<512, DD, 0, s>>>(x, stats, NNODES, prerelu);
    bn_finalize<<<1, DD, 0, s>>>(stats, NNODES);
    const long n = (long)NNODES * DD;
    bn_apply<<<(int)((n + 255) / 256), 256, 0, s>>>(x, y, stats, n, prerelu);
}

extern "C" void kernel_launch(void* const* d_in, const int* in_sizes, int n_in,
                              void* d_out, int out_size, void* d_ws, size_t ws_size,
                              hipStream_t stream)
{
    (void)in_sizes; (void)n_in; (void)out_size; (void)ws_size;

    const float* f_in  = (const float*)d_in[0];
    const int*   erow  = (const int*)  d_in[1];
    const int*   ecol  = (const int*)  d_in[2];
    const float* vgat  = (const float*)d_in[3];
    const float* vutt  = (const float*)d_in[4];
    const float* vpas  = (const float*)d_in[5];
    const float* vfut  = (const float*)d_in[6];
    const float* gWs   = (const float*)d_in[7];   // [L,H,128,16]
    const float* gbs   = (const float*)d_in[8];   // [L,H,16]
    const float* gWn   = (const float*)d_in[9];
    const float* gbn   = (const float*)d_in[10];
    const float* gas   = (const float*)d_in[11];  // [L,H,16]
    const float* gan   = (const float*)d_in[12];
    const float* Wut   = (const float*)d_in[13];  // [L,128,128]
    const float* but   = (const float*)d_in[14];  // [L,128]
    const float* Wpa   = (const float*)d_in[15];
    const float* bpa   = (const float*)d_in[16];
    const float* Wfu   = (const float*)d_in[17];
    const float* bfu   = (const float*)d_in[18];
    const float* Wsf   = (const float*)d_in[19];  // [128,128]
    const float* bsf   = (const float*)d_in[20];  // [128]
    const float* Wcls  = (const float*)d_in[21];  // [640,7]
    const float* bcls  = (const float*)d_in[22];  // [7]
    float* out = (float*)d_out;                   // [N,7]

    // ---- workspace layout (floats); ~234 MB total ----
    const size_t NF = (size_t)NNODES * DD;
    float* ws      = (float*)d_ws;
    float* o_gat   = ws + 0 * NF;
    float* o_uttr  = ws + 1 * NF;
    float* o_past  = ws + 2 * NF;
    float* o_futr  = ws + 3 * NF;
    float* o_self  = ws + 4 * NF;
    float* tmpF    = ws + 5 * NF;   // features between layers
    float* tmpH1   = ws + 6 * NF;   // hs / GCN h
    float* tmpH2   = ws + 7 * NF;   // hn
    float* tmpAgg  = ws + 8 * NF;   // scatter destination
    float* atsBuf  = ws + 9 * NF;
    float* atnBuf  = atsBuf + (size_t)NNODES * NH;
    float* stats   = atnBuf + (size_t)NNODES * NH;  // 256 floats

    const int E = NEDGES;
    const int scatterBlocks = (E * 32 + 255) / 256;
    const int attBlocks     = (NNODES * NH + 255) / 256;
    const int fillBlocks    = (int)((NF + 255) / 256);

    // ---------------- self branch: bn(relu(f_in @ Wsf + bsf)) ----------------
    gemm_full(f_in, Wsf, bsf, tmpH1, /*relu=*/true, stream);
    run_bn(tmpH1, o_self, stats, /*prerelu=*/0, stream);

    // ---------------- GAT branch ----------------
    for (int l = 0; l < NLAY; ++l) {
        const float* f  = (l == 0) ? f_in : tmpF;
        const float* Ws = gWs + (size_t)l * NH * DD * DHH;
        const float* Wn = gWn + (size_t)l * NH * DD * DHH;
        const float* bs = gbs + (size_t)l * NH * DHH;
        const float* bn = gbn + (size_t)l * NH * DHH;
        gemm_head(f, Ws, bs, tmpH1, stream);   // hs = relu(f @ Ws + bs)
        gemm_head(f, Wn, bn, tmpH2, stream);   // hn = relu(f @ Wn + bn)
        gat_att<<<attBlocks, 256, 0, stream>>>(tmpH1,
                                               gas + (size_t)l * NH * DHH,
                                               gan + (size_t)l * NH * DHH,
                                               atsBuf, atnBuf, NNODES);
        fillf<<<fillBlocks, 256, 0, stream>>>(tmpAgg, (long)NF, 0.0f);
        scatter_edges<true><<<scatterBlocks, 256, 0, stream>>>(tmpH2, atsBuf, atnBuf,
                                                               erow, ecol, vgat,
                                                               tmpAgg, E);
        run_bn(tmpAgg, (l == NLAY - 1) ? o_gat : tmpF, stats, /*prerelu=*/0, stream);
    }

    // ---------------- three GCN branches ----------------
    const float* Wb[3] = {Wut, Wpa, Wfu};
    const float* bb[3] = {but, bpa, bfu};
    const float* vb[3] = {vutt, vpas, vfut};
    float*       ob[3] = {o_uttr, o_past, o_futr};
    for (int br = 0; br < 3; ++br) {
        for (int l = 0; l < NLAY; ++l) {
            const float* f = (l == 0) ? f_in : tmpF;
            gemm_full(f, Wb[br] + (size_t)l * DD * DD, bb[br] + (size_t)l * DD,
                      tmpH1, /*relu=*/false, stream);           // h = f@W + b
            fillf<<<fillBlocks, 256, 0, stream>>>(tmpAgg, (long)NF, 0.0f);
            scatter_edges<false><<<scatterBlocks, 256, 0, stream>>>(tmpH1, nullptr, nullptr,
                                                                    erow, ecol, vb[br],
                                                                    tmpAgg, E);
            run_bn(tmpAgg, (l == NLAY - 1) ? ob[br] : tmpF, stats, /*prerelu=*/1, stream);
        }
    }

    // ---------------- classifier: concat @ Wcls + bcls as 5 accumulating GEMMs
    gemm_cls(o_gat,  Wcls + 0 * DD * NCLS, bcls, out, /*first=*/true,  stream);
    gemm_cls(o_uttr, Wcls + 1 * DD * NCLS, nullptr, out, /*first=*/false, stream);
    gemm_cls(o_past, Wcls + 2 * DD * NCLS, nullptr, out, /*first=*/false, stream);
    gemm_cls(o_futr, Wcls + 3 * DD * NCLS, nullptr, out, /*first=*/false, stream);
    gemm_cls(o_self, Wcls + 4 * DD * NCLS, nullptr, out, /*first=*/false, stream);
}